// SetToGraphTri_17351667876266
// MI455X (gfx1250) — compile-verified
//
#include <hip/hip_runtime.h>

typedef __attribute__((ext_vector_type(2))) float v2f;
typedef __attribute__((ext_vector_type(8))) float v8f;

#define BATCH 8
#define NROW  1024
#define FDIM  10
#define KPAD  12
#define H1    128
#define H2    128
#define HOUT  32

// -------- WMMA helper: D = A(16x4) * B(4x16) + C, fp32 --------
__device__ __forceinline__ v8f wmma_f32_4(v2f a, v2f b, v8f c) {
    return __builtin_amdgcn_wmma_f32_16x16x4_f32(false, a, false, b, (short)0, c, false, false);
}

// ================= Kernel 1: fused 3-layer MLP =================
// One wave (32 threads) per 16-row tile of the flattened (8192 x 10) input.
__global__ __launch_bounds__(32) void mlp_kernel(
    const float* __restrict__ x,
    const float* __restrict__ W0, const float* __restrict__ b0,
    const float* __restrict__ W1, const float* __restrict__ b1,
    const float* __restrict__ W2, const float* __restrict__ b2,
    float* __restrict__ h_out, float* __restrict__ sq_out)
{
    __shared__ __align__(16) float xs[16][KPAD];     // x tile, K padded with zeros
    __shared__ __align__(16) float w0t[H1][KPAD];    // W0 transposed, K padded
    __shared__ __align__(16) float w2t[HOUT][H2];    // W2 transposed
    __shared__ __align__(16) float h0[16][H1];
    __shared__ __align__(16) float h1[16][H2];
    __shared__ __align__(16) float ht[16][HOUT];

    const int  lane   = threadIdx.x;
    const int  m      = lane & 15;          // A-matrix row for this lane
    const int  koff   = (lane >> 4) * 2;    // K offset within 4-chunk
    const int  nlo    = lane & 15;          // B/C column within tile
    const int  rowoff = (lane >> 4) * 8;    // C row offset
    const long m0     = (long)blockIdx.x * 16;

    // ---- stage x tile (16 x 10, contiguous 160 floats) with zero pad ----
    for (int i = lane; i < 16 * FDIM; i += 32)
        xs[i / FDIM][i % FDIM] = x[m0 * FDIM + i];
    if (lane < 16) { xs[lane][10] = 0.f; xs[lane][11] = 0.f; }
    // ---- stage W0 transposed (coalesced read), zero pad rows 10..11 ----
    for (int i = lane; i < FDIM * H1; i += 32)
        w0t[i % H1][i / H1] = W0[i];
    for (int i = lane; i < H1 * 2; i += 32)
        w0t[i >> 1][10 + (i & 1)] = 0.f;
    // ---- stage W2 transposed ----
    for (int i = lane; i < H2 * HOUT; i += 32)
        w2t[i % HOUT][i / HOUT] = W2[i];
    __syncthreads();

    // ---- Layer 0: (16x12) @ (12x128) from LDS ----
    for (int nt = 0; nt < 8; ++nt) {
        const int nbase = nt * 16;
        v8f acc = {};
        for (int kk = 0; kk < 3; ++kk) {
            const int k = kk * 4 + koff;
            const v2f a = *(const v2f*)&xs[m][k];
            const v2f b = *(const v2f*)&w0t[nbase + nlo][k];
            acc = wmma_f32_4(a, b, acc);
        }
        for (int r = 0; r < 8; ++r) {
            const int row = r + rowoff;
            const int col = nbase + nlo;
            const float v = acc[r] + b0[col];
            h0[row][col] = v > 0.f ? v : 0.f;       // ReLU
        }
    }
    __syncthreads();

    // ---- Layer 1: (16x128) @ (128x128); B from global (L2-resident) ----
    for (int nt = 0; nt < 8; ++nt) {
        const int nbase = nt * 16;
        v8f acc = {};
        for (int kk = 0; kk < 32; ++kk) {
            const int k = kk * 4 + koff;
            const v2f a = *(const v2f*)&h0[m][k];
            v2f b;
            b.x = W1[k       * H2 + nbase + nlo];
            b.y = W1[(k + 1) * H2 + nbase + nlo];
            acc = wmma_f32_4(a, b, acc);
        }
        for (int r = 0; r < 8; ++r) {
            const int row = r + rowoff;
            const int col = nbase + nlo;
            const float v = acc[r] + b1[col];
            h1[row][col] = v > 0.f ? v : 0.f;       // ReLU
        }
    }
    __syncthreads();

    // ---- Layer 2: (16x128) @ (128x32), B from transposed LDS, no ReLU ----
    for (int nt = 0; nt < 2; ++nt) {
        const int nbase = nt * 16;
        v8f acc = {};
        for (int kk = 0; kk < 32; ++kk) {
            const int k = kk * 4 + koff;
            const v2f a = *(const v2f*)&h1[m][k];
            const v2f b = *(const v2f*)&w2t[nbase + nlo][k];
            acc = wmma_f32_4(a, b, acc);
        }
        for (int r = 0; r < 8; ++r) {
            const int row = r + rowoff;
            const int col = nbase + nlo;
            const float v = acc[r] + b2[col];
            ht[row][col] = v;
            h_out[(m0 + row) * HOUT + col] = v;
        }
    }
    __syncthreads();

    // ---- per-row squared norms ----
    if (lane < 16) {
        float s = 0.f;
        for (int c = 0; c < HOUT; ++c) s += ht[lane][c] * ht[lane][c];
        sq_out[m0 + lane] = s;
    }
}

// ============ Kernel 2: per-batch "all gt equal" flag ============
__global__ __launch_bounds__(256) void single_kernel(
    const int* __restrict__ gt, unsigned* __restrict__ flags)
{
    const int b     = blockIdx.x >> 6;
    const int chunk = blockIdx.x & 63;
    const int4* g4 = (const int4*)(gt + (size_t)b * NROW * NROW);
    const int ref = gt[(size_t)b * NROW * NROW];
    const int elems4 = NROW * NROW / 4 / 64;       // int4 per chunk
    const size_t base = (size_t)chunk * elems4;
    bool ok = true;
    for (int i = threadIdx.x; i < elems4; i += blockDim.x) {
        __builtin_prefetch(&g4[base + i + 2 * 256], 0, 0);
        const int4 v = g4[base + i];
        ok &= (v.x == ref) & (v.y == ref) & (v.z == ref) & (v.w == ref);
    }
    __shared__ int sok;
    if (threadIdx.x == 0) sok = 1;
    __syncthreads();
    if (!ok) atomicAnd(&sok, 0);
    __syncthreads();
    if (threadIdx.x == 0) atomicAnd(flags + b, sok ? ~0u : 0u);
}

// ===== Kernel 3: 16x16 Gram tile via WMMA -> thresholded graphs =====
__global__ __launch_bounds__(32) void dists_kernel(
    const float* __restrict__ h, const float* __restrict__ sq,
    float* __restrict__ graphs)
{
    __shared__ __align__(16) float hi[16][HOUT];
    __shared__ __align__(16) float hj[16][HOUT];

    const int b  = blockIdx.z;
    const int i0 = blockIdx.y * 16;
    const int j0 = blockIdx.x * 16;
    const float* hb  = h  + (size_t)b * NROW * HOUT;
    const float* sqb = sq + (size_t)b * NROW;
    float*       gb  = graphs + (size_t)b * NROW * NROW;

    const int lane   = threadIdx.x;
    const int koff   = (lane >> 4) * 2;
    const int nlo    = lane & 15;
    const int rowoff = (lane >> 4) * 8;

    // stage both 16x32 h tiles with coalesced b128 loads
    {
        const float4* si = (const float4*)(hb + (size_t)i0 * HOUT);
        const float4* sj = (const float4*)(hb + (size_t)j0 * HOUT);
        float4* di = (float4*)hi;
        float4* dj = (float4*)hj;
        for (int i = lane; i < 16 * HOUT / 4; i += 32) {
            di[i] = si[i];
            dj[i] = sj[i];
        }
    }
    __syncthreads();

    v8f acc = {};
    for (int kk = 0; kk < 8; ++kk) {
        const int k = kk * 4 + koff;
        const v2f a  = *(const v2f*)&hi[nlo][k];   // A[m][k], m = lane&15
        const v2f bb = *(const v2f*)&hj[nlo][k];   // B[k][n] = h[j0+n][k]
        acc = wmma_f32_4(a, bb, acc);
    }
    for (int r = 0; r < 8; ++r) {
        const int row = i0 + r + rowoff;
        const int col = j0 + nlo;
        const float d = sqb[row] + sqb[col] - 2.f * acc[r];
        gb[(size_t)row * NROW + col] = (d <= 1.0f) ? 0.5f : -0.5f;
    }
}

// ================ Kernel 4: triplet margin loss ================
__global__ __launch_bounds__(256) void triplet_kernel(
    const int* __restrict__ gt, const int* __restrict__ tri,
    const float* __restrict__ h, const float* __restrict__ sq,
    const unsigned* __restrict__ flags, float* __restrict__ loss_out,
    int ntri)
{
    const int b = blockIdx.x;
    const int*   gb  = gt + (size_t)b * NROW * NROW;
    const float* hb  = h  + (size_t)b * NROW * HOUT;
    const float* sqb = sq + (size_t)b * NROW;
    const bool single = (flags[b] != 0u);

    float vsum = 0.f;
    int   csum = 0;
    for (int t = threadIdx.x; t < ntri; t += blockDim.x) {
        const int a = tri[t * 3 + 0];
        const int p = tri[t * 3 + 1];
        const int n = tri[t * 3 + 2];
        const bool valid = (a != p) && (a != n) && (p != n);
        const bool pos = (gb[(size_t)a * NROW + p] != 0) &&
                         (gb[(size_t)a * NROW + n] == 0);
        const bool sel = pos && valid && !single;
        if (sel) {
            const float* ha = hb + (size_t)a * HOUT;
            const float* hp = hb + (size_t)p * HOUT;
            const float* hn = hb + (size_t)n * HOUT;
            float dotp = 0.f, dotn = 0.f;
            for (int c = 0; c < HOUT; ++c) {
                dotp += ha[c] * hp[c];
                dotn += ha[c] * hn[c];
            }
            const float d_ap = sqb[a] + sqb[p] - 2.f * dotp;
            const float d_an = sqb[a] + sqb[n] - 2.f * dotn;
            const float v = d_ap - d_an + 2.0f;   // MARGIN
            vsum += (v > 0.f) ? v : 0.f;
            csum += 1;
        }
    }
    __shared__ float svals[256];
    __shared__ int   scnt[256];
    svals[threadIdx.x] = vsum;
    scnt[threadIdx.x]  = csum;
    __syncthreads();
    for (int s = 128; s > 0; s >>= 1) {
        if (threadIdx.x < (unsigned)s) {
            svals[threadIdx.x] += svals[threadIdx.x + s];
            scnt[threadIdx.x]  += scnt[threadIdx.x + s];
        }
        __syncthreads();
    }
    if (threadIdx.x == 0) {
        const float lb = scnt[0] > 0 ? svals[0] / (float)scnt[0] : 0.f;
        atomicAdd(loss_out, lb);
    }
}

extern "C" void kernel_launch(void* const* d_in, const int* in_sizes, int n_in,
                              void* d_out, int out_size, void* d_ws, size_t ws_size,
                              hipStream_t stream) {
    const float* x   = (const float*)d_in[0];
    const int*   gt  = (const int*)  d_in[1];
    const int*   tri = (const int*)  d_in[2];
    const float* W0  = (const float*)d_in[3];
    const float* b0  = (const float*)d_in[4];
    const float* W1  = (const float*)d_in[5];
    const float* b1  = (const float*)d_in[6];
    const float* W2  = (const float*)d_in[7];
    const float* b2  = (const float*)d_in[8];

    float* graphs = (float*)d_out;
    float* loss   = graphs + (size_t)BATCH * NROW * NROW;

    // workspace: h (8192x32 f32), sq (8192 f32), flags (8 u32)
    float*    h     = (float*)d_ws;
    float*    sq    = h + (size_t)BATCH * NROW * HOUT;
    unsigned* flags = (unsigned*)(sq + (size_t)BATCH * NROW);

    hipMemsetAsync(flags, 0xFF, BATCH * sizeof(unsigned), stream);
    hipMemsetAsync(loss, 0, sizeof(float), stream);

    mlp_kernel<<<BATCH * NROW / 16, 32, 0, stream>>>(x, W0, b0, W1, b1, W2, b2, h, sq);
    single_kernel<<<BATCH * 64, 256, 0, stream>>>(gt, flags);
    dim3 dg(NROW / 16, NROW / 16, BATCH);
    dists_kernel<<<dg, 32, 0, stream>>>(h, sq, graphs);
    const int ntri = in_sizes[2] / 3;
    triplet_kernel<<<BATCH, 256, 0, stream>>>(gt, tri, h, sq, flags, loss, ntri);
}